// FM_Layer_15101105013325
// MI455X (gfx1250) — compile-verified
//
#include <hip/hip_runtime.h>

// FM layer fused kernel for MI455X (gfx1250, wave32).
// dense: [4096,13] f32, sparse: [4096,26] i32, w0:[1], w:[2600013], V:[16,2600013]
// out: [4096] f32

typedef __attribute__((ext_vector_type(2))) float v2f;
typedef __attribute__((ext_vector_type(8))) float v8f;

#define NBATCH 4096
#define ND     13
#define NF     26
#define PERF   100000
#define FEATN  (NF * PERF + ND)   /* 2600013 */
#define NWAVES 8                  /* waves per block (field-parallel) */

__global__ __launch_bounds__(32 * NWAVES)
void fm_fused_kernel(const float* __restrict__ dense,
                     const int*   __restrict__ sp,
                     const float* __restrict__ w0,
                     const float* __restrict__ w,
                     const float* __restrict__ V,
                     float*       __restrict__ out)
{
    const int tid    = threadIdx.x;
    const int lane   = tid & 31;
    const int wid    = tid >> 5;           // 0..7
    const int lane16 = lane & 15;          // k index (N dim of WMMA D)
    const int h      = lane >> 4;          // half: which 8-row group this lane owns
    const int base   = blockIdx.x * 16;    // batch-row tile base

    // Sparse index tile, preloaded coalesced; value = global embedding row.
    __shared__ int   lidx[16 * NF];                 // 1664 B
    __shared__ float ls[NWAVES - 1][32][24];        // 21504 B partials

    for (int t = tid; t < 16 * NF; t += 32 * NWAVES)
        lidx[t] = sp[base * NF + t] + ND + (t % NF) * PERF;

    // Per-lane accumulators: rows (base + 8h + j), latent dim k = lane16.
    float e[8], sq[8], wa[8];
#pragma unroll
    for (int j = 0; j < 8; ++j) { e[j] = 0.f; sq[j] = 0.f; wa[j] = 0.f; }

    if (wid == 0) {
        // ---- Dense contribution via V_WMMA_F32_16X16X4_F32, K = 13 padded to 16 ----
        // A (16x4): lane%16 = M (batch row), VGPRs hold K = 4*kc + 2*(lane/16) + {0,1}
        // B (4x16): lane%16 = N (k dim),    same K striping as A
        // D (16x16 f32): VGPR j = M offset j (+8 for lanes 16-31), lane%16 = N
        v8f ce = {};
        v8f cs = {};
#pragma unroll
        for (int kc = 0; kc < 4; ++kc) {
            const int j0   = kc * 4 + 2 * h;
            const int rowA = (base + lane16) * ND;
            float a0 = (j0     < ND) ? dense[rowA + j0]     : 0.f;
            float a1 = (j0 + 1 < ND) ? dense[rowA + j0 + 1] : 0.f;
            const size_t vrow = (size_t)lane16 * (size_t)FEATN;
            float b0 = (j0     < ND) ? V[vrow + j0]     : 0.f;
            float b1 = (j0 + 1 < ND) ? V[vrow + j0 + 1] : 0.f;
            v2f a  = {a0, a1};
            v2f b  = {b0, b1};
            v2f a2 = {a0 * a0, a1 * a1};
            v2f b2 = {b0 * b0, b1 * b1};
            ce = __builtin_amdgcn_wmma_f32_16x16x4_f32(false, a,  false, b,
                                                       (short)0, ce, false, false);
            cs = __builtin_amdgcn_wmma_f32_16x16x4_f32(false, a2, false, b2,
                                                       (short)0, cs, false, false);
        }
#pragma unroll
        for (int j = 0; j < 8; ++j) { e[j] = ce[j]; sq[j] = cs[j]; }

        // Dense first-order term, distributed over k lanes (collapsed by the
        // final 16-lane butterfly): lane k accumulates dense[row,k]*w[k].
        if (lane16 < ND) {
            const float wk = w[lane16];
#pragma unroll
            for (int j = 0; j < 8; ++j)
                wa[j] = dense[(base + 8 * h + j) * ND + lane16] * wk;
        }
    }

    __syncthreads();   // lidx ready

    // ---- Sparse gather phase: fields split over the 8 waves ----
    // Residue mapping gives wave 0 (which also did WMMA) a 3-field class;
    // waves 6,7 take the 4-field classes.
    const float* Vk = V + (size_t)lane16 * (size_t)FEATN;  // this lane's k-row of V
    for (int f = (wid + 2) & 7; f < NF; f += NWAVES) {
#pragma unroll
        for (int j = 0; j < 8; ++j) {
            const int   c = lidx[(8 * h + j) * NF + f];    // LDS broadcast read
            const float v = Vk[c];
            e[j] += v;
            sq[j] = fmaf(v, v, sq[j]);
            // exactly one lane per (row, field) folds in the first-order weight
            if ((f & 15) == lane16) wa[j] += w[c];
        }
    }

    // ---- Combine wave partials (must merge e before squaring) ----
    if (wid > 0) {
#pragma unroll
        for (int j = 0; j < 8; ++j) {
            ls[wid - 1][lane][j]      = e[j];
            ls[wid - 1][lane][8 + j]  = sq[j];
            ls[wid - 1][lane][16 + j] = wa[j];
        }
    }
    __syncthreads();

    if (wid == 0) {
#pragma unroll
        for (int wv = 0; wv < NWAVES - 1; ++wv) {
#pragma unroll
            for (int j = 0; j < 8; ++j) {
                e[j]  += ls[wv][lane][j];
                sq[j] += ls[wv][lane][8 + j];
                wa[j] += ls[wv][lane][16 + j];
            }
        }
        const float w0v = w0[0];
#pragma unroll
        for (int j = 0; j < 8; ++j) {
            // 0.5*(e_k^2 - sq_k) + first-order share; butterfly over the 16
            // k-lanes of this half (masks 8,4,2,1 stay within the half).
            float r = 0.5f * (e[j] * e[j] - sq[j]) + wa[j];
            r += __shfl_xor(r, 8, 32);
            r += __shfl_xor(r, 4, 32);
            r += __shfl_xor(r, 2, 32);
            r += __shfl_xor(r, 1, 32);
            if (lane16 == 0) out[base + 8 * h + j] = w0v + r;
        }
    }
}

extern "C" void kernel_launch(void* const* d_in, const int* in_sizes, int n_in,
                              void* d_out, int out_size, void* d_ws, size_t ws_size,
                              hipStream_t stream) {
    (void)in_sizes; (void)n_in; (void)out_size; (void)d_ws; (void)ws_size;
    const float* dense = (const float*)d_in[0];
    const int*   sp    = (const int*)d_in[1];
    const float* w0    = (const float*)d_in[2];
    const float* w     = (const float*)d_in[3];
    const float* V     = (const float*)d_in[4];
    float* out = (float*)d_out;

    fm_fused_kernel<<<dim3(NBATCH / 16), dim3(32 * NWAVES), 0, stream>>>(
        dense, sp, w0, w, V, out);
}